// SoftTreeEnsembleLayer_85822036509455
// MI455X (gfx1250) — compile-verified
//
#include <hip/hip_runtime.h>
#include <hip/hip_bf16.h>

typedef __attribute__((ext_vector_type(2))) float v2f;
typedef __attribute__((ext_vector_type(4))) float v4f;
typedef __attribute__((ext_vector_type(8))) float v8f;

#define N_ 4096
#define F_ 256
#define T_ 64
#define S_ 15
#define L_ 16
#define K_ 32
#define O_ 128
#define NTILE 16
#define TCHUNK 8
#define WSTR 36   // staged-W k-stride: 32 feats + bias + 3 zero pads (even -> b64-aligned reads)

__launch_bounds__(256)
__global__ void soft_tree_wmma_kernel(const float* __restrict__ x,
                                      const float* __restrict__ split_coefs,
                                      const int*   __restrict__ leaves_feat_ids,
                                      const float* __restrict__ leaves_coefs,
                                      float*       __restrict__ out)
{
    __shared__ float lds_x[NTILE * F_];               // 16 KB: 16 x-rows
    __shared__ float lds_route[TCHUNK * L_ * NTILE];  //  8 KB: routing weights, one t-chunk
    __shared__ float lds_w[L_ * NTILE * WSTR];        // 36 KB: staged A-operand for one tree

    const int tid = threadIdx.x;
    const int n0  = blockIdx.x * NTILE;

    // ---- Stage x tile into LDS (contiguous 16x256 block, coalesced)
    for (int i = tid; i < NTILE * F_; i += 256)
        lds_x[i] = x[n0 * F_ + i];
    __syncthreads();

    // ---- WMMA lane identity (A: M=jj | B/D: N=jj ; lanes>=16 hold K+2,K+3)
    const int wave = tid >> 5;
    const int lane = tid & 31;
    const int jj   = lane & 15;
    const int kh   = lane >> 4;
    const int o0   = wave * 16;

    // ---- W-build identity: one (leaf, n) row per thread (256 = 16*16)
    const int b_l  = tid >> 4;
    const int b_nl = tid & 15;

    v8f acc = {};

    for (int tc = 0; tc < T_ / TCHUNK; ++tc) {
        // ============ Phase 1: routing for this t-chunk (threads 0..127) ============
        if (tid < 128) {
            const int r_nl   = tid & 15;
            const int r_tloc = tid >> 4;            // 0..7
            const int t = tc * TCHUNK + r_tloc;

            float dots[S_];
            #pragma unroll
            for (int s = 0; s < S_; ++s)
                dots[s] = split_coefs[(t * S_ + s) * (F_ + 1) + F_];   // bias

            const float* xr = &lds_x[r_nl * F_];
            #pragma unroll
            for (int s = 0; s < S_; ++s) {
                const float* sc = &split_coefs[(t * S_ + s) * (F_ + 1)];
                float a = dots[s];
                for (int f = 0; f < F_; ++f)
                    a = fmaf(xr[f], sc[f], a);
                dots[s] = a;
            }
            float sst[S_];
            #pragma unroll
            for (int s = 0; s < S_; ++s) {
                float tcv = fminf(0.5f, fmaxf(-0.5f, dots[s]));
                sst[s] = tcv * (1.5f - 2.0f * tcv * tcv) + 0.5f;
            }
            float a[L_];
            a[0] = 1.0f;
            #pragma unroll
            for (int d = 0; d < 4; ++d) {
                const int off = (1 << d) - 1;
                #pragma unroll
                for (int i = (1 << d) - 1; i >= 0; --i) {
                    const float sv = sst[off + i];
                    const float av = a[i];
                    a[2 * i + 1] = av * sv;
                    a[2 * i]     = av * (1.0f - sv);
                }
            }
            #pragma unroll
            for (int l = 0; l < L_; ++l)
                lds_route[(r_tloc * L_ + l) * NTILE + r_nl] = a[l];
        }
        __syncthreads();

        // ============ Phase 2: per tree, cooperative W-stage then WMMA GEMM ============
        for (int tloc = 0; tloc < TCHUNK; ++tloc) {
            const int t = tc * TCHUNK + tloc;

            __syncthreads();   // previous tree's GEMM finished reading lds_w
            // ---- Build W[l][n][k] = route * x[n, fid[k]]  (+ bias col, + zero pad)
            {
                const float r   = lds_route[(tloc * L_ + b_l) * NTILE + b_nl];
                const int*  fid = &leaves_feat_ids[(t * L_ + b_l) * K_];
                const float* xr = &lds_x[b_nl * F_];
                float* wrow = &lds_w[(b_l * NTILE + b_nl) * WSTR];
                #pragma unroll
                for (int c = 0; c < 8; ++c) {
                    const int4 f4 = ((const int4*)fid)[c];
                    v4f w;
                    w.x = r * xr[f4.x];
                    w.y = r * xr[f4.y];
                    w.z = r * xr[f4.z];
                    w.w = r * xr[f4.w];
                    *(v4f*)&wrow[c * 4] = w;
                }
                v4f bw;
                bw.x = r; bw.y = 0.0f; bw.z = 0.0f; bw.w = 0.0f;
                *(v4f*)&wrow[K_] = bw;          // k=32 bias, k=33..35 zeros
            }
            __syncthreads();

            // ---- GEMM: 16 leaves x 9 k-chunks of f32 WMMA (K=4 each)
            const float* cb = &leaves_coefs[(size_t)(t * L_) * O_ * (K_ + 1)];
            for (int l = 0; l < L_; ++l) {
                const float* wrow = &lds_w[(l * NTILE + jj) * WSTR];
                const float* cbo  = &cb[((size_t)l * O_ + o0 + jj) * (K_ + 1)];
                #pragma unroll
                for (int c = 0; c < 8; ++c) {
                    const int kbase = c * 4 + kh * 2;
                    v2f av = *(const v2f*)&wrow[kbase];   // (r*x[f0], r*x[f1])
                    v2f bv;
                    bv.x = cbo[kbase];
                    bv.y = cbo[kbase + 1];
                    acc = __builtin_amdgcn_wmma_f32_16x16x4_f32(
                        false, av, false, bv, (short)0, acc, false, false);
                }
                // Bias chunk: A-side holds (r, 0) on kh=0 lanes and (0, 0) on kh=1
                // lanes (zero-padded W), so B can be loaded uniformly on ALL lanes:
                // kh=1 lanes' B values multiply a guaranteed 0.  bv.y stays 0 so no
                // uninitialized value can create 0*NaN.
                {
                    v2f av = *(const v2f*)&wrow[K_ + kh * 2];  // (r,0) | (0,0)
                    v2f bv;
                    bv.x = cbo[K_];
                    bv.y = 0.0f;
                    acc = __builtin_amdgcn_wmma_f32_16x16x4_f32(
                        false, av, false, bv, (short)0, acc, false, false);
                }
            }
        }
    }

    // ---- Store 16x16 tile (C/D layout: lane<16 -> M=v, lane>=16 -> M=v+8)
    #pragma unroll
    for (int v = 0; v < 8; ++v)
        out[(size_t)(n0 + v + kh * 8) * O_ + (o0 + jj)] = acc[v];
}

extern "C" void kernel_launch(void* const* d_in, const int* in_sizes, int n_in,
                              void* d_out, int out_size, void* d_ws, size_t ws_size,
                              hipStream_t stream) {
    const float* x               = (const float*)d_in[0];
    const float* split_coefs     = (const float*)d_in[1];
    const int*   leaves_feat_ids = (const int*)d_in[2];
    const float* leaves_coefs    = (const float*)d_in[3];
    float* out = (float*)d_out;

    dim3 grid(N_ / NTILE);   // 256 workgroups (16 n-rows each)
    dim3 block(256);         // 8 wave32: waves tile O=128 in 16-col slabs
    soft_tree_wmma_kernel<<<grid, block, 0, stream>>>(
        x, split_coefs, leaves_feat_ids, leaves_coefs, out);
}